// AutoGraderPrototypeModel_4252017623198
// MI455X (gfx1250) — compile-verified
//
#include <hip/hip_runtime.h>
#include <hip/hip_bf16.h>

// Problem constants (from the reference)
#define BB   1024   // batch
#define TT   256    // time
#define DD   1024   // hidden dim
#define MM   512    // prototypes total
#define NL   8      // labels
#define NP   64     // prototypes per label

typedef __attribute__((ext_vector_type(2))) float v2f;
typedef __attribute__((ext_vector_type(4))) float f4;
typedef __attribute__((ext_vector_type(8))) float v8f;

// ---------------------------------------------------------------------------
// Kernel 1: streaming mean-pool over T  (the 1 GB / ~46us bandwidth-bound part)
// One block per batch row. 256 threads x float4 = exactly one D-row per t-step.
// Non-temporal loads: input has zero reuse and is 5x the L2 — keep it out.
// Fused: per-row squared norm a_sq[b] via LDS block reduction.
// ---------------------------------------------------------------------------
__global__ __launch_bounds__(256) void pool_kernel(const float* __restrict__ hs,
                                                   float* __restrict__ a,
                                                   float* __restrict__ a_sq) {
  const int b   = blockIdx.x;
  const int tid = threadIdx.x;
  const float* row = hs + (size_t)b * TT * DD + (size_t)tid * 4;

  f4 acc = {0.f, 0.f, 0.f, 0.f};
#pragma unroll 4
  for (int t = 0; t < TT; ++t) {
    f4 v = __builtin_nontemporal_load((const f4*)(row + (size_t)t * DD));
    acc += v;
  }
  acc *= (1.0f / (float)TT);

  // pooled row is hot (GEMM A operand) -> regular temporal store, stays in L2
  *(f4*)(a + (size_t)b * DD + (size_t)tid * 4) = acc;

  // fused ||a_b||^2
  float ss = acc.x * acc.x + acc.y * acc.y + acc.z * acc.z + acc.w * acc.w;
  __shared__ float red[256];
  red[tid] = ss;
  __syncthreads();
#pragma unroll
  for (int s = 128; s > 0; s >>= 1) {
    if (tid < s) red[tid] += red[tid + s];
    __syncthreads();
  }
  if (tid == 0) a_sq[b] = red[0];
}

// ---------------------------------------------------------------------------
// Kernel 2: prototype squared norms b_sq[m]  (2 MB read, trivial)
// ---------------------------------------------------------------------------
__global__ __launch_bounds__(256) void bsq_kernel(const float* __restrict__ w,
                                                  float* __restrict__ b_sq) {
  const int m   = blockIdx.x;
  const int tid = threadIdx.x;
  f4 v = *(const f4*)(w + (size_t)m * DD + (size_t)tid * 4);
  float ss = v.x * v.x + v.y * v.y + v.z * v.z + v.w * v.w;
  __shared__ float red[256];
  red[tid] = ss;
  __syncthreads();
#pragma unroll
  for (int s = 128; s > 0; s >>= 1) {
    if (tid < s) red[tid] += red[tid + s];
    __syncthreads();
  }
  if (tid == 0) b_sq[m] = red[0];
}

// ---------------------------------------------------------------------------
// Kernel 3: ab = a (1024x1024) * w^T (1024x512) via V_WMMA_F32_16X16X4_F32.
// One 16x16 output tile per wave; 2048 waves = 256 blocks x 8 waves.
// A fragment (16x4 f32, ISA layout): lane l holds row m0+(l&15), k = kb..kb+1,
// kb = (l>>4)*2  -> one contiguous 8B (b64) load per operand per k-step.
// B fragment (4x16): lane l holds col n0+(l&15) -> w[col][kb..kb+1] contiguous.
// Operands are L2-resident (6 MB total), so direct global feeds are fine.
// EXEC is all-ones (full 256-thread blocks, no divergence) as WMMA requires.
// ---------------------------------------------------------------------------
__global__ __launch_bounds__(256) void wmma_gemm_kernel(const float* __restrict__ A,
                                                        const float* __restrict__ W,
                                                        float* __restrict__ ab) {
  const int wave = (blockIdx.x * 256 + threadIdx.x) >> 5;  // 0..2047
  const int lane = threadIdx.x & 31;
  const int mt   = wave >> 5;   // 0..63   (M tiles)
  const int nt   = wave & 31;   // 0..31   (N tiles)

  const int ln15 = lane & 15;
  const int kb   = (lane >> 4) << 1;          // 0 or 2
  const float* ap = A + (size_t)((mt << 4) + ln15) * DD + kb;
  const float* bp = W + (size_t)((nt << 4) + ln15) * DD + kb;

  v8f acc = {};
#pragma unroll 8
  for (int k0 = 0; k0 < DD; k0 += 4) {
    v2f af = *(const v2f*)(ap + k0);
    v2f bf = *(const v2f*)(bp + k0);
    // D = A*B + C  (fp32, RNE) — emits v_wmma_f32_16x16x4_f32
    acc = __builtin_amdgcn_wmma_f32_16x16x4_f32(
        /*neg_a=*/false, af, /*neg_b=*/false, bf,
        /*c_mod=*/(short)0, acc, /*reuse_a=*/false, /*reuse_b=*/false);
  }

  // C/D layout: lane l, VGPR v -> (M = v + (l>>4)*8, N = l&15)
  const int mrow0 = (mt << 4) + ((lane >> 4) << 3);
  float* outp = ab + (size_t)mrow0 * MM + (nt << 4) + ln15;
#pragma unroll
  for (int v = 0; v < 8; ++v) outp[(size_t)v * MM] = acc[v];
}

// ---------------------------------------------------------------------------
// Kernel 4: fold norm-expansion + quirky (B, 64, 8).mean(axis=1) epilogue.
// out[b][l] = (2/64)*sum_p ab[b, p*8+l] - a_sq[b] - (1/64)*sum_p b_sq[p*8+l]
// idx = tid + 64*i keeps l = tid&7 fixed per thread and is fully coalesced.
// ---------------------------------------------------------------------------
__global__ __launch_bounds__(64) void finalize_kernel(const float* __restrict__ ab,
                                                      const float* __restrict__ a_sq,
                                                      const float* __restrict__ b_sq,
                                                      float* __restrict__ out) {
  const int b   = blockIdx.x;
  const int tid = threadIdx.x;  // 64 threads = 8 labels x 8 groups
  float s_ab = 0.f, s_bq = 0.f;
#pragma unroll
  for (int i = 0; i < 8; ++i) {
    const int idx = tid + 64 * i;           // idx % 8 == tid % 8 == label
    s_ab += ab[(size_t)b * MM + idx];
    s_bq += b_sq[idx];
  }
  __shared__ float r_ab[64];
  __shared__ float r_bq[64];
  r_ab[tid] = s_ab;
  r_bq[tid] = s_bq;
  __syncthreads();
  if (tid < NL) {
    float sa = 0.f, sb = 0.f;
#pragma unroll
    for (int g = 0; g < 8; ++g) {
      sa += r_ab[g * 8 + tid];
      sb += r_bq[g * 8 + tid];
    }
    out[(size_t)b * NL + tid] =
        (2.0f / (float)NP) * sa - a_sq[b] - (1.0f / (float)NP) * sb;
  }
}

// ---------------------------------------------------------------------------
extern "C" void kernel_launch(void* const* d_in, const int* in_sizes, int n_in,
                              void* d_out, int out_size, void* d_ws, size_t ws_size,
                              hipStream_t stream) {
  const float* hs = (const float*)d_in[0];   // [B, T, D] f32
  const float* w  = (const float*)d_in[1];   // [M, D]    f32
  float* out      = (float*)d_out;           // [B, NL]   f32

  char* ws = (char*)d_ws;
  float* a_mean = (float*)ws;                                   // B*D   = 4 MB
  float* ab     = (float*)(ws + (size_t)BB * DD * sizeof(float));  // B*M = 2 MB
  float* a_sq   = (float*)(ws + (size_t)BB * DD * sizeof(float)
                              + (size_t)BB * MM * sizeof(float));  // B
  float* b_sq   = a_sq + BB;                                       // M

  // 1) bandwidth-dominant streaming mean-pool (+ fused a_sq)
  pool_kernel<<<BB, 256, 0, stream>>>(hs, a_mean, a_sq);
  // 2) prototype norms
  bsq_kernel<<<MM, 256, 0, stream>>>(w, b_sq);
  // 3) fp32 WMMA GEMM: 64 x 32 tiles of 16x16, one per wave
  wmma_gemm_kernel<<<(BB / 16) * (MM / 16) / 8, 256, 0, stream>>>(a_mean, w, ab);
  // 4) epilogue / label-mean reduction
  finalize_kernel<<<BB, 64, 0, stream>>>(ab, a_sq, b_sq, out);
}